// POLICY_41334765256938
// MI455X (gfx1250) — compile-verified
//
#include <hip/hip_runtime.h>
#include <hip/hip_bf16.h>

// ---------------------------------------------------------------------------
// GIN (3 layers, H=64) + MLP head for MI455X (gfx1250, wave32, WMMA).
//  - edge scatter-add: L2-resident f32 atomics (global_atomic_add_f32)
//  - MLP GEMMs: exact fp32 via V_WMMA_F32_16X16X4_F32
//  - A-tile staging via gfx1250 async global->LDS copy (ASYNCcnt)
//  - BN batch stats fused into the GEMM kernel (LDS ds_add_f32 + global atomics)
//  - BN-apply fused with next layer's (1+eps)*x init (saves one N*H pass)
// ---------------------------------------------------------------------------

#define HID 64
#define WSTRIDE 68   // LDS row stride (floats) to avoid 64-bank conflicts
#define BN_EPS 1e-5f

typedef __attribute__((ext_vector_type(2))) float v2f;
typedef __attribute__((ext_vector_type(8))) float v8f;

// ---------------------------------------------------------------------------
// Layer 0 only: agg = (1 + eps[0]) * x
// ---------------------------------------------------------------------------
__global__ __launch_bounds__(256)
void gin_init_agg(const float* __restrict__ x, float* __restrict__ agg,
                  const float* __restrict__ eps_arr, int layer, int N)
{
    int t = blockIdx.x * 256 + threadIdx.x;           // one float4 per thread
    int total = N * (HID / 4);
    if (t < total) {
        float e = 1.0f + eps_arr[layer];              // uniform scalar load
        float4 v = *(const float4*)(x + (size_t)t * 4);
        v.x *= e; v.y *= e; v.z *= e; v.w *= e;
        *(float4*)(agg + (size_t)t * 4) = v;
    }
}

// ---------------------------------------------------------------------------
// agg[dst] += x[src] over E edges. 16 threads per edge (float4 chunks),
// 16 edges per 256-thread block; src/dst staged in LDS (coalesced index load).
// Block 0 also zeroes the 128-float BN stats buffer for the following MLP.
// ---------------------------------------------------------------------------
__global__ __launch_bounds__(256)
void gin_scatter(const int* __restrict__ ei, long long E,
                 const float* __restrict__ x, float* __restrict__ agg,
                 float* __restrict__ stats)
{
    __shared__ int ss[16], sd[16];
    long long e0 = (long long)blockIdx.x * 16;
    int t = threadIdx.x;
    if (blockIdx.x == 0 && t >= 128) stats[t - 128] = 0.0f;
    if (t < 16) {
        long long e = e0 + t;
        ss[t] = (e < E) ? ei[e] : 0;
    } else if (t < 32) {
        long long e = e0 + (t - 16);
        sd[t - 16] = (e < E) ? ei[E + e] : 0;
    }
    __syncthreads();

    int el = t >> 4;           // edge within block
    int q  = t & 15;           // float4 chunk within row
    if (e0 + el < E) {
        int s = ss[el];
        int d = sd[el];
        float4 v = *(const float4*)(x + (size_t)s * HID + q * 4);
        float* dst = agg + (size_t)d * HID + q * 4;
        __hip_atomic_fetch_add(dst + 0, v.x, __ATOMIC_RELAXED, __HIP_MEMORY_SCOPE_AGENT);
        __hip_atomic_fetch_add(dst + 1, v.y, __ATOMIC_RELAXED, __HIP_MEMORY_SCOPE_AGENT);
        __hip_atomic_fetch_add(dst + 2, v.z, __ATOMIC_RELAXED, __HIP_MEMORY_SCOPE_AGENT);
        __hip_atomic_fetch_add(dst + 3, v.w, __ATOMIC_RELAXED, __HIP_MEMORY_SCOPE_AGENT);
    }
}

// ---------------------------------------------------------------------------
// h = relu(relu(agg @ W1^T + b1) @ W2^T + b2), fused BN batch-stat reduction.
// 8 waves/block, each wave owns a 16-row tile; WMMA f32 16x16x4.
// A tile staged with gfx1250 async global->LDS copies.
// ---------------------------------------------------------------------------
__global__ __launch_bounds__(256)
void gin_mlp(const float* __restrict__ agg,
             const float* __restrict__ W1, const float* __restrict__ b1,
             const float* __restrict__ W2, const float* __restrict__ b2,
             float* __restrict__ hout, float* __restrict__ stats, int N)
{
    __shared__ float W1s[HID * WSTRIDE];
    __shared__ float W2s[HID * WSTRIDE];
    __shared__ float b1s[HID], b2s[HID];
    __shared__ float As[8][16 * WSTRIDE];   // per-wave A tile (reused for h1)
    __shared__ float ssum[HID], ssq[HID];

    const int t = threadIdx.x;
    if (t < HID) { ssum[t] = 0.0f; ssq[t] = 0.0f; }
    for (int i = t; i < HID * HID; i += 256) {
        int r = i >> 6, c = i & 63;
        W1s[r * WSTRIDE + c] = W1[i];
        W2s[r * WSTRIDE + c] = W2[i];
    }
    if (t < HID) { b1s[t] = b1[t]; b2s[t] = b2[t]; }

    const int wave = t >> 5;
    const int lane = t & 31;
    const int m  = lane & 15;       // row (A) / col (B,C)
    const int hi = lane >> 4;       // half-wave select
    const int kh = hi * 2;          // K sub-offset for A/B fragments
    const int rowBase = blockIdx.x * 128 + wave * 16;
    float* Aw = As[wave];

    // --- stage 16x64 agg tile into LDS via async global->LDS (ASYNCcnt) ---
    // Generic LDS pointer: low 32 bits = wave-relative LDS byte offset.
    {
        unsigned ldsBase = (unsigned)(unsigned long long)(void*)Aw;
        for (int i = 0; i < 8; ++i) {
            int rl  = i * 2 + hi;
            int row = rowBase + rl;
            if (row < N) {   // OOB tail rows skipped: stale LDS only feeds
                             // guarded OOB output rows (excluded from stats)
                unsigned lds = ldsBase + (unsigned)((rl * WSTRIDE + m * 4) * 4);
                unsigned long long ga =
                    (unsigned long long)(agg + (size_t)row * HID + m * 4);
                asm volatile("global_load_async_to_lds_b128 %0, %1, off"
                             :: "v"(lds), "v"(ga) : "memory");
            }
        }
        asm volatile("s_wait_asynccnt 0" ::: "memory");
    }
    __syncthreads();   // weights + biases + stats zero visible block-wide

    // --- GEMM1: h1 = relu(A @ W1^T + b1) ---
    float h1v[4][8];
    for (int jb = 0; jb < 4; ++jb) {
        v8f c = {};
        for (int kk = 0; kk < HID; kk += 4) {
            v2f a = *(const v2f*)(Aw + m * WSTRIDE + kk + kh);
            v2f b = *(const v2f*)(W1s + (jb * 16 + m) * WSTRIDE + kk + kh);
            c = __builtin_amdgcn_wmma_f32_16x16x4_f32(false, a, false, b,
                                                      (short)0, c, false, false);
        }
        float bias = b1s[jb * 16 + m];
        for (int r = 0; r < 8; ++r)
            h1v[jb][r] = fmaxf(c[r] + bias, 0.0f);
    }
    // restage h1 into the same LDS tile (per-wave LDS is hardware in-order)
    for (int jb = 0; jb < 4; ++jb) {
        int col = jb * 16 + m;
        for (int r = 0; r < 8; ++r)
            Aw[(r + 8 * hi) * WSTRIDE + col] = h1v[jb][r];
    }

    // --- GEMM2: h2 = relu(h1 @ W2^T + b2), store + BN stats ---
    for (int jb = 0; jb < 4; ++jb) {
        v8f c = {};
        for (int kk = 0; kk < HID; kk += 4) {
            v2f a = *(const v2f*)(Aw + m * WSTRIDE + kk + kh);
            v2f b = *(const v2f*)(W2s + (jb * 16 + m) * WSTRIDE + kk + kh);
            c = __builtin_amdgcn_wmma_f32_16x16x4_f32(false, a, false, b,
                                                      (short)0, c, false, false);
        }
        int col = jb * 16 + m;
        float bias = b2s[col];
        float s1 = 0.0f, s2 = 0.0f;
        for (int r = 0; r < 8; ++r) {
            int row = rowBase + r + 8 * hi;
            float v = fmaxf(c[r] + bias, 0.0f);
            if (row < N) {
                hout[(size_t)row * HID + col] = v;
                s1 += v; s2 += v * v;
            }
        }
        __hip_atomic_fetch_add(&ssum[col], s1, __ATOMIC_RELAXED, __HIP_MEMORY_SCOPE_WORKGROUP);
        __hip_atomic_fetch_add(&ssq[col],  s2, __ATOMIC_RELAXED, __HIP_MEMORY_SCOPE_WORKGROUP);
    }
    __syncthreads();
    if (t < HID)
        __hip_atomic_fetch_add(&stats[t], ssum[t], __ATOMIC_RELAXED, __HIP_MEMORY_SCOPE_AGENT);
    else if (t < 2 * HID)
        __hip_atomic_fetch_add(&stats[t], ssq[t - HID], __ATOMIC_RELAXED, __HIP_MEMORY_SCOPE_AGENT);
}

// ---------------------------------------------------------------------------
// Fused: y = BN(h) written in place, and agg = (1+eps[next])*y for next layer.
// ---------------------------------------------------------------------------
__global__ __launch_bounds__(256)
void gin_bn_fuse(float* __restrict__ h, float* __restrict__ agg,
                 const float* __restrict__ stats,
                 const float* __restrict__ gamma, const float* __restrict__ beta,
                 const float* __restrict__ eps_arr, int next_layer, int N)
{
    int t = blockIdx.x * 256 + threadIdx.x;
    int total = N * (HID / 4);
    if (t >= total) return;
    int q = t & 15;
    float invN = 1.0f / (float)N;
    float e = 1.0f + eps_arr[next_layer];
    float4 v = *(const float4*)(h + (size_t)t * 4);
    float4 w;
    float* vp = (float*)&v;
    float* wp = (float*)&w;
#pragma unroll
    for (int i = 0; i < 4; ++i) {
        int f = q * 4 + i;
        float mean  = stats[f] * invN;
        float var   = stats[HID + f] * invN - mean * mean;
        float scale = gamma[f] * rsqrtf(var + BN_EPS);
        float y     = (vp[i] - mean) * scale + beta[f];
        vp[i] = y;
        wp[i] = e * y;
    }
    *(float4*)(h + (size_t)t * 4) = v;
    *(float4*)(agg + (size_t)t * 4) = w;
}

// ---------------------------------------------------------------------------
// Last layer: in-place BN only.
// ---------------------------------------------------------------------------
__global__ __launch_bounds__(256)
void gin_bn_apply(float* __restrict__ h, const float* __restrict__ stats,
                  const float* __restrict__ gamma, const float* __restrict__ beta,
                  int N)
{
    int t = blockIdx.x * 256 + threadIdx.x;
    int total = N * (HID / 4);
    if (t >= total) return;
    int q = t & 15;
    float invN = 1.0f / (float)N;
    float4 v = *(float4*)(h + (size_t)t * 4);
    float* vp = (float*)&v;
#pragma unroll
    for (int i = 0; i < 4; ++i) {
        int f = q * 4 + i;
        float mean = stats[f] * invN;
        float var  = stats[HID + f] * invN - mean * mean;
        vp[i] = gamma[f] * (vp[i] - mean) * rsqrtf(var + BN_EPS) + beta[f];
    }
    *(float4*)(h + (size_t)t * 4) = v;
}

// ---------------------------------------------------------------------------
// out[n] = lin2_w . relu(x[n] @ lin1_w^T + lin1_b) + lin2_b
// ---------------------------------------------------------------------------
__global__ __launch_bounds__(256)
void gin_head(const float* __restrict__ x,
              const float* __restrict__ lin1w, const float* __restrict__ lin1b,
              const float* __restrict__ lin2w, const float* __restrict__ lin2b,
              float* __restrict__ out, int N)
{
    __shared__ float Ws[HID * WSTRIDE];
    __shared__ float bs[HID], w2s[HID];
    __shared__ float As[8][16 * WSTRIDE];

    const int t = threadIdx.x;
    for (int i = t; i < HID * HID; i += 256)
        Ws[(i >> 6) * WSTRIDE + (i & 63)] = lin1w[i];
    if (t < HID) { bs[t] = lin1b[t]; w2s[t] = lin2w[t]; }
    __syncthreads();

    const int wave = t >> 5;
    const int lane = t & 31;
    const int m  = lane & 15;
    const int hi = lane >> 4;
    const int kh = hi * 2;
    const int rowBase = blockIdx.x * 128 + wave * 16;
    float* Aw = As[wave];

    for (int i = 0; i < 8; ++i) {
        int rl = i * 2 + hi;
        int row = rowBase + rl;
        float4 v = make_float4(0.f, 0.f, 0.f, 0.f);
        if (row < N) v = *(const float4*)(x + (size_t)row * HID + m * 4);
        *(float4*)(Aw + rl * WSTRIDE + m * 4) = v;
    }

    float acc[8] = {0.f, 0.f, 0.f, 0.f, 0.f, 0.f, 0.f, 0.f};
    for (int jb = 0; jb < 4; ++jb) {
        v8f c = {};
        for (int kk = 0; kk < HID; kk += 4) {
            v2f a = *(const v2f*)(Aw + m * WSTRIDE + kk + kh);
            v2f b = *(const v2f*)(Ws + (jb * 16 + m) * WSTRIDE + kk + kh);
            c = __builtin_amdgcn_wmma_f32_16x16x4_f32(false, a, false, b,
                                                      (short)0, c, false, false);
        }
        int col = jb * 16 + m;
        float bias = bs[col];
        float w2 = w2s[col];
#pragma unroll
        for (int r = 0; r < 8; ++r)
            acc[r] += fmaxf(c[r] + bias, 0.0f) * w2;
    }
    // reduce over the 16 lanes of each half-wave (cols 0..63 total)
    for (int off = 8; off >= 1; off >>= 1)
#pragma unroll
        for (int r = 0; r < 8; ++r)
            acc[r] += __shfl_xor(acc[r], off, 16);
    if (m == 0) {
        float b2v = lin2b[0];
#pragma unroll
        for (int r = 0; r < 8; ++r) {
            int row = rowBase + r + 8 * hi;
            if (row < N) out[row] = acc[r] + b2v;
        }
    }
}

// ---------------------------------------------------------------------------
extern "C" void kernel_launch(void* const* d_in, const int* in_sizes, int n_in,
                              void* d_out, int out_size, void* d_ws, size_t ws_size,
                              hipStream_t stream)
{
    const float* x      = (const float*)d_in[0];
    const int*   ei     = (const int*)d_in[1];
    const float* W1     = (const float*)d_in[2];
    const float* b1     = (const float*)d_in[3];
    const float* W2     = (const float*)d_in[4];
    const float* b2     = (const float*)d_in[5];
    const float* gamma  = (const float*)d_in[6];
    const float* beta   = (const float*)d_in[7];
    const float* eps    = (const float*)d_in[8];
    const float* lin1w  = (const float*)d_in[9];
    const float* lin1b  = (const float*)d_in[10];
    const float* lin2w  = (const float*)d_in[11];
    const float* lin2b  = (const float*)d_in[12];
    float* out = (float*)d_out;

    const int       N = in_sizes[0] / HID;
    const long long E = (long long)in_sizes[1] / 2;

    float* agg   = (float*)d_ws;                   // [N, 64]
    float* h     = agg + (size_t)N * HID;          // [N, 64]
    float* stats = h + (size_t)N * HID;            // [128] (sum, sumsq)

    const int gridElem = (N * (HID / 4) + 255) / 256;
    const int gridEdge = (int)((E + 15) / 16);
    const int gridRows = (N + 127) / 128;

    gin_init_agg<<<gridElem, 256, 0, stream>>>(x, agg, eps, 0, N);
    for (int l = 0; l < 3; ++l) {
        const float* xin = (l == 0) ? x : h;   // h holds normalized x after fuse
        gin_scatter<<<gridEdge, 256, 0, stream>>>(ei, E, xin, agg, stats);
        gin_mlp    <<<gridRows, 256, 0, stream>>>(agg, W1 + l * HID * HID, b1 + l * HID,
                                                  W2 + l * HID * HID, b2 + l * HID,
                                                  h, stats, N);
        if (l < 2)
            gin_bn_fuse<<<gridElem, 256, 0, stream>>>(h, agg, stats, gamma + l * HID,
                                                      beta + l * HID, eps, l + 1, N);
        else
            gin_bn_apply<<<gridElem, 256, 0, stream>>>(h, stats, gamma + l * HID,
                                                       beta + l * HID, N);
    }
    gin_head<<<gridRows, 256, 0, stream>>>(h, lin1w, lin1b, lin2w, lin2b, out, N);
}